// PreProcesser_47021301957489
// MI455X (gfx1250) — compile-verified
//
#include <hip/hip_runtime.h>
#include <hip/hip_bf16.h>

// CDNA5 / gfx1250 implementation.
// One workgroup (256 threads = 8 wave32) per residue (Z*N = 512).
// Phase 1: backbone + frames (scalar, thread 0) -> LDS + global out.
// Phase 2: local_voxels via V_WMMA_F32_16X16X4_F32 (homogeneous coords:
//          [gx gy gz 1] x [x_unit; y_unit; z_unit; origin]), then 14-atom
//          field accumulation + per-point normalization -> LDS.
// Phase 3: divergence (all components differentiated along the VX axis,
//          exactly as the reference's _div_component slices axis 2).

typedef __attribute__((ext_vector_type(2))) float v2f;
typedef __attribute__((ext_vector_type(8))) float v8f;

#define N_RES   512
#define N_ATOM  14
#define NPTS    4096   // 16*16*16

__global__ __launch_bounds__(256)
void field_div_kernel(const float* __restrict__ C,          // (512,14,3)
                      const int*   __restrict__ L,          // (512)
                      const float* __restrict__ atom_mask,  // (512,14)
                      const float* __restrict__ amber,      // (21,14)
                      float* __restrict__ out_cb,           // (512,4,3)
                      float* __restrict__ out_div,          // (512,4096)
                      float* __restrict__ out_frames)       // (512,3,3)
{
    __shared__ float sF[3][3];        // frames[u][s]: rows x_unit,y_unit,z_unit
    __shared__ float sOrg[3];         // origin = cb
    __shared__ float sC[N_ATOM][3];   // atom coords
    __shared__ float sQ[N_ATOM];      // charges
    __shared__ float FX[NPTS];
    __shared__ float FY[NPTS];
    __shared__ float FZ[NPTS];

    const int res = blockIdx.x;
    const int tid = threadIdx.x;
    const float* Cr = C + res * N_ATOM * 3;

    // ---------------- Phase 1: backbone / frames (thread 0) ----------------
    if (tid == 0) {
        float n0 = Cr[0], n1 = Cr[1], n2 = Cr[2];
        float a0 = Cr[3], a1 = Cr[4], a2 = Cr[5];   // ca
        float c0 = Cr[6], c1 = Cr[7], c2 = Cr[8];
        float b10 = a0 - n0, b11 = a1 - n1, b12 = a2 - n2;          // ca - n
        float b20 = c0 - a0, b21 = c1 - a1, b22 = c2 - a2;          // c - ca
        float b30 = b11*b22 - b12*b21;                               // cross(b1,b2)
        float b31 = b12*b20 - b10*b22;
        float b32 = b10*b21 - b11*b20;
        float cb0 = a0 - 0.58273431f*b20 + 0.56802827f*b10 - 0.54067466f*b30;
        float cb1 = a1 - 0.58273431f*b21 + 0.56802827f*b11 - 0.54067466f*b31;
        float cb2 = a2 - 0.58273431f*b22 + 0.56802827f*b12 - 0.54067466f*b32;

        // y = cb - ca, normalized
        float y0 = cb0 - a0, y1 = cb1 - a1, y2 = cb2 - a2;
        float yn = fmaxf(sqrtf(y0*y0 + y1*y1 + y2*y2), 1e-6f);
        float yu0 = y0/yn, yu1 = y1/yn, yu2 = y2/yn;

        // x_raw = c - n;  x = x_raw - dot(x_raw, yu)   (scalar broadcast, per reference)
        float xr0 = c0 - n0, xr1 = c1 - n1, xr2 = c2 - n2;
        float xp  = xr0*yu0 + xr1*yu1 + xr2*yu2;
        float x0 = xr0 - xp, x1 = xr1 - xp, x2 = xr2 - xp;
        float xn = fmaxf(sqrtf(x0*x0 + x1*x1 + x2*x2), 1e-6f);
        float xu0 = x0/xn, xu1 = x1/xn, xu2 = x2/xn;

        // z = cross(x_unit, y_unit)
        float zu0 = xu1*yu2 - xu2*yu1;
        float zu1 = xu2*yu0 - xu0*yu2;
        float zu2 = xu0*yu1 - xu1*yu0;

        sF[0][0]=xu0; sF[0][1]=xu1; sF[0][2]=xu2;
        sF[1][0]=yu0; sF[1][1]=yu1; sF[1][2]=yu2;
        sF[2][0]=zu0; sF[2][1]=zu1; sF[2][2]=zu2;
        sOrg[0]=cb0; sOrg[1]=cb1; sOrg[2]=cb2;

        float* cbp = out_cb + res * 12;
        cbp[0]=n0;  cbp[1]=n1;  cbp[2]=n2;
        cbp[3]=a0;  cbp[4]=a1;  cbp[5]=a2;
        cbp[6]=c0;  cbp[7]=c1;  cbp[8]=c2;
        cbp[9]=cb0; cbp[10]=cb1; cbp[11]=cb2;

        float* fr = out_frames + res * 9;
        fr[0]=xu0; fr[1]=xu1; fr[2]=xu2;
        fr[3]=yu0; fr[4]=yu1; fr[5]=yu2;
        fr[6]=zu0; fr[7]=zu1; fr[8]=zu2;
    }
    if (tid < N_ATOM * 3) sC[tid / 3][tid % 3] = Cr[tid];
    if (tid < N_ATOM) {
        int lbl = L[res];
        if (lbl == -1) lbl = 20;                       // X_LBL
        sQ[tid] = amber[lbl * N_ATOM + tid] * atom_mask[res * N_ATOM + tid];
    }
    __syncthreads();

    // ---------------- Phase 2: WMMA local voxels + field sum ----------------
    const int  lane = tid & 31;
    const int  wave = tid >> 5;
    const int  n16  = lane & 15;
    const bool hi   = lane >= 16;

    // A (16x4 f32): row m<3 = [x_unit[m], y_unit[m], z_unit[m], origin[m]]^T slots.
    // Layout: lanes 0-15 hold A[m][0],A[m][1]; lanes 16-31 hold A[m][2],A[m][3].
    v2f aOp;
    {
        int m = hi ? (lane - 16) : lane;
        float ax = 0.f, ay = 0.f;
        if (m < 3) {
            ax = hi ? sF[2][m] : sF[0][m];   // K=2 row -> z_unit ; K=0 row -> x_unit
            ay = hi ? sOrg[m]  : sF[1][m];   // K=3 row -> origin ; K=1 row -> y_unit
        }
        aOp.x = ax; aOp.y = ay;
    }

    const v8f czero = {0.f,0.f,0.f,0.f,0.f,0.f,0.f,0.f};

    for (int it = 0; it < 16; ++it) {
        const int base = wave * 512 + it * 32;
        const int p0 = base + n16;          // tile0 column for this lane
        const int p1 = p0 + 16;             // tile1 column

        // voxel coords: p = x*256 + y*16 + v;  gx=x-8, gy=y-4, gz=v-8 (RES=1)
        float g0x = (float)(p0 >> 8)        - 8.f;
        float g0y = (float)((p0 >> 4) & 15) - 4.f;
        float g0z = (float)(p0 & 15)        - 8.f;
        float g1x = (float)(p1 >> 8)        - 8.f;
        float g1y = (float)((p1 >> 4) & 15) - 4.f;
        float g1z = (float)(p1 & 15)        - 8.f;

        // B (4x16 f32): lane n holds column n; lo lanes = K rows 0,1 (gx,gy),
        // hi lanes = K rows 2,3 (gz, 1)
        v2f b0, b1;
        b0.x = hi ? g0z : g0x;  b0.y = hi ? 1.f : g0y;
        b1.x = hi ? g1z : g1x;  b1.y = hi ? 1.f : g1y;

        // D[s][n] = local point n, component s (rows 3..15 are zero)
        v8f d0 = __builtin_amdgcn_wmma_f32_16x16x4_f32(
            false, aOp, false, b0, (short)0, czero, false, false);
        v8f d1 = __builtin_amdgcn_wmma_f32_16x16x4_f32(
            false, aOp, false, b1, (short)0, czero, false, false);

        // tile1 results live in lanes 0-15; move them up to lanes 16-31
        float t1x = __shfl(d1[0], n16, 32);
        float t1y = __shfl(d1[1], n16, 32);
        float t1z = __shfl(d1[2], n16, 32);
        float px = hi ? t1x : d0[0];
        float py = hi ? t1y : d0[1];
        float pz = hi ? t1z : d0[2];

        // 14-atom field accumulation
        float fx = 0.f, fy = 0.f, fz = 0.f;
        #pragma unroll
        for (int a = 0; a < N_ATOM; ++a) {
            float dx = px - sC[a][0];
            float dy = py - sC[a][1];
            float dz = pz - sC[a][2];
            float r2   = dx*dx + dy*dy + dz*dz;
            float dist = sqrtf(r2);
            float t    = fmaxf(dist, 1.0f);                    // max(dist, RES)
            float dn   = ((dist == 0.f) ? 1.f : dist) * t * t; // denom
            float w    = sQ[a] / dn;
            fx += w * dx; fy += w * dy; fz += w * dz;
        }
        float fn  = sqrtf(fx*fx + fy*fy + fz*fz);
        float inv = 1.f / ((fn == 0.f) ? 1.f : fn);

        int p = base + lane;                // every lane owns point base+lane
        FX[p] = fx * inv;
        FY[p] = fy * inv;
        FZ[p] = fz * inv;
    }
    __syncthreads();

    // ---------------- Phase 3: divergence (all derivatives along VX) --------
    float* dv = out_div + res * NPTS;
    for (int p = tid; p < NPTS; p += 256) {
        const int x = p >> 8;               // VX index; stride 256 in flat p
        float ddx, ddy, ddz;
        if (x == 0) {
            ddx = FX[p + 256] - FX[p];
            ddy = FY[p + 256] - FY[p];
            ddz = FZ[p + 256] - FZ[p];
        } else if (x == 15) {
            ddx = FX[p]       - FX[p - 256];   // flip_last = False
            ddy = FY[p - 256] - FY[p];         // flip_last = True
            ddz = FZ[p - 256] - FZ[p];         // flip_last = True
        } else {
            ddx = 0.5f * (FX[p + 256] - FX[p - 256]);
            ddy = 0.5f * (FY[p + 256] - FY[p - 256]);
            ddz = 0.5f * (FZ[p + 256] - FZ[p - 256]);
        }
        dv[p] = ddx + ddy + ddz;
    }
}

extern "C" void kernel_launch(void* const* d_in, const int* in_sizes, int n_in,
                              void* d_out, int out_size, void* d_ws, size_t ws_size,
                              hipStream_t stream) {
    (void)in_sizes; (void)n_in; (void)d_ws; (void)ws_size; (void)out_size;
    const float* C         = (const float*)d_in[0];   // (2,256,14,3)
    const int*   L         = (const int*)  d_in[1];   // (2,256)
    const float* atom_mask = (const float*)d_in[2];   // (2,256,14)
    // d_in[3] = valid_mask (unused by the reference)
    const float* amber     = (const float*)d_in[4];   // (21,14)

    float* out        = (float*)d_out;
    float* out_cb     = out;                               // 512*12   = 6144
    float* out_div    = out + N_RES * 12;                  // 512*4096 = 2097152
    float* out_frames = out + N_RES * 12 + N_RES * NPTS;   // 512*9    = 4608

    field_div_kernel<<<N_RES, 256, 0, stream>>>(C, L, atom_mask, amber,
                                                out_cb, out_div, out_frames);
}